// SRUCell_75754633167423
// MI455X (gfx1250) — compile-verified
//
#include <hip/hip_runtime.h>

typedef __attribute__((ext_vector_type(4))) float    v4f;
typedef __attribute__((ext_vector_type(4))) unsigned v4u;
typedef __attribute__((ext_vector_type(8))) unsigned v8u;
typedef __attribute__((ext_vector_type(4))) int      v4i;
typedef __attribute__((ext_vector_type(8))) int      v8i;

// ---- fast tanh (prefer gfx1250 v_tanh_f32 if the builtin exists) ----
__device__ __forceinline__ float fast_tanh(float x) {
#if __has_builtin(__builtin_amdgcn_tanhf)
  return __builtin_amdgcn_tanhf(x);
#else
  float e = __builtin_amdgcn_exp2f(x * 2.8853900817779268f);  // 2*log2(e)
  return 1.0f - 2.0f * __builtin_amdgcn_rcpf(e + 1.0f);
#endif
}

#define CFENCE() asm volatile("" ::: "memory")

#if __has_builtin(__builtin_amdgcn_s_wait_tensorcnt)
#define WAIT_TENSOR(n) do { __builtin_amdgcn_s_wait_tensorcnt((short)(n)); CFENCE(); } while (0)
#else
#define WAIT_TENSOR(n) asm volatile("s_wait_tensorcnt %0" :: "n"(n) : "memory")
#endif

// Problem constants (reference): B=16, T=2048, U=1024
constexpr int Bc  = 16;
constexpr int Tc  = 2048;
constexpr int Uc  = 1024;
constexpr int X3U = 3 * Uc;                    // 3072 floats per (b,t) row
constexpr int NT  = 32;                        // one wave per block
constexpr int VEC = 4;                         // channels per lane
constexpr int CPB = NT * VEC;                  // 128 channels per block
constexpr int BLOCKS_PER_B = Uc / CPB;         // 8

constexpr int TCH    = 8;                      // timesteps per TDM descriptor
constexpr int NCHUNK = 5;                      // descriptors in flight / LDS slots
constexpr int NCHUNKS_TOT = Tc / TCH;          // 256
constexpr int ROWS = 3 * TCH;                  // 24 rows (z|f|r per step)
constexpr int CHUNK_FLOATS = ROWS * CPB;       // 3072 floats = 12 KB per chunk
constexpr unsigned long long CHUNK_GBYTES = (unsigned long long)TCH * X3U * 4;  // 98304

// ---- one TDM descriptor: 2D tile, 24 rows x 512B, row pitch 4KB ----
__device__ __forceinline__ void tdm_issue(unsigned lds_off, unsigned long long gaddr) {
  v4u g0;
  g0.x = 1u;                                            // count=1, user mode, no gather
  g0.y = lds_off;                                       // lds_addr (bytes)
  g0.z = (unsigned)gaddr;                               // global_addr[31:0]
  g0.w = ((unsigned)(gaddr >> 32) & 0x01FFFFFFu)        // global_addr[56:32]
         | (2u << 30);                                  // type=2 ("image")
  v8u g1;
  g1.s0 = 0x00020000u;                                  // data_size=2 (4B); mask=0; no pad/iter
  g1.s1 = (128u & 0xFFFFu) << 16;                       // tensor_dim0 lo16 (=128 floats)
  g1.s2 = ((unsigned)ROWS & 0xFFFFu) << 16;             // tensor_dim0 hi16=0 | tensor_dim1 lo16=24
  g1.s3 = (128u << 16);                                 // tensor_dim1 hi16=0 | tile_dim0=128
  g1.s4 = (unsigned)ROWS;                               // tile_dim1=24 | tile_dim2=0 (2D)
  g1.s5 = 1024u;                                        // tensor_dim0_stride lo32 (floats)
  g1.s6 = 0u;                                           // stride hi16 | dim1_stride lo16
  g1.s7 = 0u;                                           // dim1_stride hi32 (unused, 2D)
#if __has_builtin(__builtin_amdgcn_tensor_load_to_lds)
  v4i zq = {0, 0, 0, 0};                                // groups 2/3 unused (tile_dim2=0)
  v8i zo = {0, 0, 0, 0, 0, 0, 0, 0};                    // extra group (unused)
  CFENCE();
  __builtin_amdgcn_tensor_load_to_lds(g0, (v8i)g1, zq, zq, zo, 0);
  CFENCE();
#else
  asm volatile("tensor_load_to_lds %0, %1" :: "s"(g0), "s"(g1) : "memory");
#endif
}

__global__ __launch_bounds__(NT)
void sru_scan_tdm_kernel(const float* __restrict__ x,   // [B, T, 3U]
                         const float* __restrict__ c0,  // [B, U]
                         float* __restrict__ hout,      // [B, T, U]
                         float* __restrict__ cout)      // [B, U]
{
  __shared__ float smem[NCHUNK * CHUNK_FLOATS];         // 5 * 12KB = 60KB

  const int tid = threadIdx.x;
  const int b   = blockIdx.x >> 3;                      // BLOCKS_PER_B == 8
  const int u0  = (blockIdx.x & (BLOCKS_PER_B - 1)) * CPB;
  const int u   = u0 + tid * VEC;

  // Low 32 bits of a flat LDS pointer == workgroup-relative LDS byte offset.
  const unsigned lds_base = (unsigned)(size_t)(void*)smem;
  const unsigned long long gbase =
      (unsigned long long)(size_t)x + ((size_t)b * Tc * X3U + u0) * 4ull;

  // ---- preload NCHUNK chunks (each = 8 timesteps of z|f|r for our slice) ----
  #pragma unroll
  for (int i = 0; i < NCHUNK; ++i)
    tdm_issue(lds_base + (unsigned)i * (CHUNK_FLOATS * 4),
              gbase + (unsigned long long)i * CHUNK_GBYTES);

  v4f c = *(const v4f*)(c0 + b * Uc + u);
  float* hb = hout + (size_t)b * Tc * Uc + u;

  auto process_chunk = [&](const float* sp, float* hp) {
    #pragma unroll
    for (int s = 0; s < TCH; ++s) {
      v4f z = *(const v4f*)(sp + s * (3 * CPB));
      v4f f = *(const v4f*)(sp + s * (3 * CPB) + CPB);
      v4f r = *(const v4f*)(sp + s * (3 * CPB) + 2 * CPB);
      c = z + f * (c - z);                    // f*c + (1-f)*z
      v4f th;
      th.x = fast_tanh(c.x); th.y = fast_tanh(c.y);
      th.z = fast_tanh(c.z); th.w = fast_tanh(c.w);
      v4f h = z + r * (th - z);               // r*tanh(c) + (1-r)*z
      __builtin_nontemporal_store(h, (v4f*)(hp + s * Uc));
    }
  };

  int slot = 0;
  unsigned long long gnext = gbase + (unsigned long long)NCHUNK * CHUNK_GBYTES;

  int ch = 0;
  // ---- steady state: wait for oldest descriptor, consume 8 steps, refill ----
  for (; ch < NCHUNKS_TOT - NCHUNK; ++ch) {
    WAIT_TENSOR(NCHUNK - 1);                  // oldest in-flight descriptor done
    process_chunk(smem + slot * CHUNK_FLOATS + tid * VEC,
                  hb + (size_t)ch * (TCH * Uc));
    tdm_issue(lds_base + (unsigned)slot * (CHUNK_FLOATS * 4), gnext);
    gnext += CHUNK_GBYTES;
    slot = (slot == NCHUNK - 1) ? 0 : slot + 1;
  }

  // ---- drain: everything already issued; wait for all, consume the rest ----
  WAIT_TENSOR(0);
  for (; ch < NCHUNKS_TOT; ++ch) {
    process_chunk(smem + slot * CHUNK_FLOATS + tid * VEC,
                  hb + (size_t)ch * (TCH * Uc));
    slot = (slot == NCHUNK - 1) ? 0 : slot + 1;
  }

  *(v4f*)(cout + b * Uc + u) = c;             // final cell state
}

extern "C" void kernel_launch(void* const* d_in, const int* in_sizes, int n_in,
                              void* d_out, int out_size, void* d_ws, size_t ws_size,
                              hipStream_t stream) {
  const float* x  = (const float*)d_in[0];    // inputs [16, 2048, 3072] f32
  const float* c0 = (const float*)d_in[1];    // c0     [16, 1024]       f32
  float* h  = (float*)d_out;                                // [16,2048,1024]
  float* cf = (float*)d_out + (size_t)Bc * Tc * Uc;         // [16,1024]

  dim3 grid(Bc * BLOCKS_PER_B);               // 128 single-wave blocks
  dim3 block(NT);                             // wave32
  sru_scan_tdm_kernel<<<grid, block, 0, stream>>>(x, c0, h, cf);
}